// MinCutTAD_33200097198467
// MI455X (gfx1250) — compile-verified
//
#include <hip/hip_runtime.h>
#include <hip/hip_bf16.h>

// ---------------------------------------------------------------------------
// MinCutTAD forward on MI455X (gfx1250, wave32, WMMA).
// Dense linear algebra goes through V_WMMA_F32_16X16X4_F32; sparse graph
// aggregation goes through global f32 atomics (memory-bound, ~200MB traffic).
// ---------------------------------------------------------------------------

typedef __attribute__((ext_vector_type(2))) float v2f;
typedef __attribute__((ext_vector_type(8))) float v8f;

#define F_IN   16
#define HID    64
#define NCLUST 100
#define NPAD   112   // 7 tiles of 16 columns for layer 2

__device__ __forceinline__ v8f wmma4(v2f a, v2f b, v8f c) {
  // D = A(16x4 f32) * B(4x16 f32) + C(16x16 f32)
  return __builtin_amdgcn_wmma_f32_16x16x4_f32(
      /*neg_a=*/false, a, /*neg_b=*/false, b,
      /*c_mod=*/(short)0, c, /*reuse_a=*/false, /*reuse_b=*/false);
}

// ----------------------- row softmax of x (width 16) -----------------------
__global__ void smc_kernel(const float* __restrict__ x, float* __restrict__ s, int n) {
  int i = blockIdx.x * blockDim.x + threadIdx.x;
  if (i >= n) return;
  float v[F_IN];
  float mx = -3.0e38f;
  #pragma unroll
  for (int k = 0; k < F_IN; ++k) { v[k] = x[i * F_IN + k]; mx = fmaxf(mx, v[k]); }
  float sum = 0.f;
  #pragma unroll
  for (int k = 0; k < F_IN; ++k) { v[k] = expf(v[k] - mx); sum += v[k]; }
  float inv = 1.f / sum;
  #pragma unroll
  for (int k = 0; k < F_IN; ++k) s[i * F_IN + k] = v[k] * inv;
}

// --------------------- edge scatter: aggr1 += ew * x[src] ------------------
__global__ void scatter16_kernel(const int* __restrict__ ei, const float* __restrict__ ew,
                                 const float* __restrict__ x, float* __restrict__ aggr,
                                 int nedges) {
  int t = blockIdx.x * blockDim.x + threadIdx.x;
  int e = t >> 4, f = t & 15;
  if (e >= nedges) return;
  int s = ei[e], d = ei[nedges + e];
  atomicAdd(&aggr[d * F_IN + f], ew[e] * x[s * F_IN + f]);
}

// --------------------- edge scatter: aggr2 += ew * h[src] ------------------
__global__ void scatter64_kernel(const int* __restrict__ ei, const float* __restrict__ ew,
                                 const float* __restrict__ h, float* __restrict__ aggr,
                                 int nedges) {
  int t = blockIdx.x * blockDim.x + threadIdx.x;
  int e = t >> 6, f = t & 63;
  if (e >= nedges) return;
  int s = ei[e], d = ei[nedges + e];
  atomicAdd(&aggr[d * HID + f], ew[e] * h[s * HID + f]);
}

// ------------------------ d_flat[src] += ew (row degree) -------------------
__global__ void dflat_kernel(const int* __restrict__ ei, const float* __restrict__ ew,
                             float* __restrict__ dflat, int nedges) {
  int e = blockIdx.x * blockDim.x + threadIdx.x;
  if (e >= nedges) return;
  atomicAdd(&dflat[ei[e]], ew[e]);
}

// ----- layer 1: h = relu(aggr1 @ W1_rel^T + x @ W1_root^T + b1), WMMA ------
__global__ __launch_bounds__(32)
void layer1_wmma(const float* __restrict__ aggr1, const float* __restrict__ x,
                 const float* __restrict__ W1_rel, const float* __restrict__ W1_root,
                 const float* __restrict__ b1, float* __restrict__ h) {
  const int lane = threadIdx.x;
  const int node_base = blockIdx.x * 16;
  const int m = lane & 15;
  const int khalf = (lane >> 4) << 1;          // 0 or 2
  const int row = node_base + m;

  v2f Aagg[4], Ax[4];
  #pragma unroll
  for (int kc = 0; kc < 4; ++kc) {
    int k = kc * 4 + khalf;
    Aagg[kc].x = aggr1[row * F_IN + k];
    Aagg[kc].y = aggr1[row * F_IN + k + 1];
    Ax[kc].x   = x[row * F_IN + k];
    Ax[kc].y   = x[row * F_IN + k + 1];
  }
  #pragma unroll
  for (int nt = 0; nt < 4; ++nt) {
    int n = nt * 16 + m;
    v8f acc = {};
    #pragma unroll
    for (int kc = 0; kc < 4; ++kc) {
      int k = kc * 4 + khalf;
      v2f Brel, Broot;                         // B[k][n] = W[n][k]
      Brel.x  = W1_rel[n * F_IN + k];
      Brel.y  = W1_rel[n * F_IN + k + 1];
      Broot.x = W1_root[n * F_IN + k];
      Broot.y = W1_root[n * F_IN + k + 1];
      acc = wmma4(Aagg[kc], Brel, acc);
      acc = wmma4(Ax[kc], Broot, acc);
    }
    float bias = b1[n];
    #pragma unroll
    for (int r = 0; r < 8; ++r) {
      int mr = r + ((lane >> 4) << 3);
      float v = acc[r] + bias;
      h[(node_base + mr) * HID + n] = v > 0.f ? v : 0.f;
    }
  }
}

// --- layer 2: s_out = aggr2 @ W2_rel^T + h @ W2_root^T + b2 (100->112) -----
__global__ __launch_bounds__(32)
void layer2_wmma(const float* __restrict__ aggr2, const float* __restrict__ h,
                 const float* __restrict__ W2_rel, const float* __restrict__ W2_root,
                 const float* __restrict__ b2, float* __restrict__ s_out) {
  const int lane = threadIdx.x;
  const int node_base = blockIdx.x * 16;
  const int m = lane & 15;
  const int khalf = (lane >> 4) << 1;
  const int row = node_base + m;

  v2f Aagg[16], Ah[16];
  #pragma unroll
  for (int kc = 0; kc < 16; ++kc) {
    int k = kc * 4 + khalf;
    Aagg[kc].x = aggr2[row * HID + k];
    Aagg[kc].y = aggr2[row * HID + k + 1];
    Ah[kc].x   = h[row * HID + k];
    Ah[kc].y   = h[row * HID + k + 1];
  }
  for (int nt = 0; nt < 7; ++nt) {
    int n = nt * 16 + m;
    float valid = (n < NCLUST) ? 1.f : 0.f;    // zero-fill padded columns
    v8f acc = {};
    #pragma unroll
    for (int kc = 0; kc < 16; ++kc) {
      int k = kc * 4 + khalf;
      int nc = (n < NCLUST) ? n : 0;
      v2f Brel, Broot;
      Brel.x  = valid * W2_rel[nc * HID + k];
      Brel.y  = valid * W2_rel[nc * HID + k + 1];
      Broot.x = valid * W2_root[nc * HID + k];
      Broot.y = valid * W2_root[nc * HID + k + 1];
      acc = wmma4(Aagg[kc], Brel, acc);
      acc = wmma4(Ah[kc], Broot, acc);
    }
    float bias = (n < NCLUST) ? b2[n] : 0.f;
    #pragma unroll
    for (int r = 0; r < 8; ++r) {
      int mr = r + ((lane >> 4) << 3);
      s_out[(node_base + mr) * NPAD + n] = acc[r] + bias;
    }
  }
}

// -------- final row softmax over the 100 valid columns -> d_out ------------
__global__ void softmax_out_kernel(const float* __restrict__ s_out,
                                   float* __restrict__ out, int n) {
  int i = blockIdx.x * blockDim.x + threadIdx.x;
  if (i >= n) return;
  const float* r = s_out + (size_t)i * NPAD;
  float mx = -3.0e38f;
  for (int j = 0; j < NCLUST; ++j) mx = fmaxf(mx, r[j]);
  float sum = 0.f;
  for (int j = 0; j < NCLUST; ++j) sum += expf(r[j] - mx);
  float inv = 1.f / sum;
  for (int j = 0; j < NCLUST; ++j) out[(size_t)i * NCLUST + j] = expf(r[j] - mx) * inv;
}

// ------ mincut numerator: sum_e ew[e] * <s_mc[src], s_mc[dst]> -------------
__global__ void cutnum_kernel(const int* __restrict__ ei, const float* __restrict__ ew,
                              const float* __restrict__ smc, float* __restrict__ acc,
                              int nedges) {
  int e = blockIdx.x * blockDim.x + threadIdx.x;
  float v = 0.f;
  if (e < nedges) {
    int s = ei[e], d = ei[nedges + e];
    float dot = 0.f;
    #pragma unroll
    for (int k = 0; k < F_IN; ++k) dot += smc[s * F_IN + k] * smc[d * F_IN + k];
    v = ew[e] * dot;
  }
  #pragma unroll
  for (int off = 16; off > 0; off >>= 1) v += __shfl_down(v, off, 32);
  if ((threadIdx.x & 31) == 0) atomicAdd(acc, v);
}

// ------ mincut denominator: sum_i dflat[i] * ||s_mc[i]||^2 -----------------
__global__ void cutden_kernel(const float* __restrict__ smc, const float* __restrict__ dflat,
                              float* __restrict__ acc, int n) {
  int i = blockIdx.x * blockDim.x + threadIdx.x;
  float v = 0.f;
  if (i < n) {
    float q = 0.f;
    #pragma unroll
    for (int k = 0; k < F_IN; ++k) { float s = smc[i * F_IN + k]; q += s * s; }
    v = dflat[i] * q;
  }
  #pragma unroll
  for (int off = 16; off > 0; off >>= 1) v += __shfl_down(v, off, 32);
  if ((threadIdx.x & 31) == 0) atomicAdd(acc, v);
}

// ------ ss = s_mc^T @ s_mc, 16x16 Gram via WMMA f32 16x16x4 ----------------
// A (= S^T chunk) and B (= S chunk) fragments coincide in lane layout.
__global__ __launch_bounds__(32)
void gram_wmma(const float* __restrict__ smc, float* __restrict__ ss,
               int rows_per_block, int n) {
  const int lane = threadIdx.x;
  const int base = blockIdx.x * rows_per_block;
  const int m = lane & 15;
  const int khalf = (lane >> 4) << 1;
  v8f acc = {};
  for (int r0 = 0; r0 < rows_per_block; r0 += 4) {
    int row = base + r0 + khalf;
    v2f a;
    a.x = (row     < n) ? smc[(row)     * F_IN + m] : 0.f;
    a.y = (row + 1 < n) ? smc[(row + 1) * F_IN + m] : 0.f;
    acc = wmma4(a, a, acc);
  }
  #pragma unroll
  for (int r = 0; r < 8; ++r) {
    int mg = r + ((lane >> 4) << 3);
    atomicAdd(&ss[mg * 16 + (lane & 15)], acc[r]);
  }
}

// --------------------- finalize: mc and ortho loss -------------------------
__global__ void finalize_kernel(const float* __restrict__ ss, const float* __restrict__ scal,
                                float* __restrict__ out_tail) {
  if (threadIdx.x != 0 || blockIdx.x != 0) return;
  float mc = -(scal[0] / scal[1]);
  float nrm2 = 0.f;
  for (int i = 0; i < 256; ++i) nrm2 += ss[i] * ss[i];
  float inv_nrm = 1.f / sqrtf(nrm2);
  const float inv_nrm_I = 0.25f;               // 1 / ||I_16||_F
  float o2 = 0.f;
  for (int i = 0; i < 16; ++i)
    for (int j = 0; j < 16; ++j) {
      float v = ss[i * 16 + j] * inv_nrm - ((i == j) ? inv_nrm_I : 0.f);
      o2 += v * v;
    }
  out_tail[0] = mc;
  out_tail[1] = sqrtf(o2);
}

// ---------------------------------------------------------------------------
extern "C" void kernel_launch(void* const* d_in, const int* in_sizes, int n_in,
                              void* d_out, int out_size, void* d_ws, size_t ws_size,
                              hipStream_t stream) {
  const float* x       = (const float*)d_in[0];
  const int*   ei      = (const int*)  d_in[1];
  const float* ew      = (const float*)d_in[2];
  const float* W1_rel  = (const float*)d_in[3];
  const float* W1_root = (const float*)d_in[4];
  const float* b1      = (const float*)d_in[5];
  const float* W2_rel  = (const float*)d_in[6];
  const float* W2_root = (const float*)d_in[7];
  const float* b2      = (const float*)d_in[8];
  float* out = (float*)d_out;

  const int N = in_sizes[0] / F_IN;      // 12000
  const int E = in_sizes[1] / 2;         // 384000

  // workspace carve-up (floats)
  float* ws     = (float*)d_ws;
  float* aggr1  = ws;                    // N*16
  float* h      = aggr1 + (size_t)N * F_IN;   // N*64
  float* aggr2  = h     + (size_t)N * HID;    // N*64
  float* s_mc   = aggr2 + (size_t)N * HID;    // N*16
  float* s_out  = s_mc  + (size_t)N * F_IN;   // N*112
  float* dflat  = s_out + (size_t)N * NPAD;   // N
  float* ss     = dflat + (size_t)N;          // 256
  float* scal   = ss + 256;                   // [0]=num [1]=den

  // zero the accumulated buffers every call (deterministic under graph replay)
  hipMemsetAsync(aggr1, 0, (size_t)N * F_IN * sizeof(float), stream);
  hipMemsetAsync(aggr2, 0, (size_t)N * HID * sizeof(float), stream);
  hipMemsetAsync(dflat, 0, ((size_t)N + 256 + 2) * sizeof(float), stream);

  const int TB = 256;

  // S = softmax(x) rows
  smc_kernel<<<(N + TB - 1) / TB, TB, 0, stream>>>(x, s_mc, N);

  // layer 1
  scatter16_kernel<<<(E * 16 + TB - 1) / TB, TB, 0, stream>>>(ei, ew, x, aggr1, E);
  layer1_wmma<<<N / 16, 32, 0, stream>>>(aggr1, x, W1_rel, W1_root, b1, h);

  // layer 2
  scatter64_kernel<<<(E * 64 + TB - 1) / TB, TB, 0, stream>>>(ei, ew, h, aggr2, E);
  layer2_wmma<<<N / 16, 32, 0, stream>>>(aggr2, h, W2_rel, W2_root, b2, s_out);

  // output softmax (N x 100)
  softmax_out_kernel<<<(N + TB - 1) / TB, TB, 0, stream>>>(s_out, out, N);

  // mincut terms
  dflat_kernel<<<(E + TB - 1) / TB, TB, 0, stream>>>(ei, ew, dflat, E);
  cutnum_kernel<<<(E + TB - 1) / TB, TB, 0, stream>>>(ei, ew, s_mc, &scal[0], E);
  cutden_kernel<<<(N + TB - 1) / TB, TB, 0, stream>>>(s_mc, dflat, &scal[1], N);

  // Gram matrix S^T S via WMMA, partial sums merged with atomics
  const int GRAM_BLOCKS = 30;
  int rows_per_block = ((N + 4 * GRAM_BLOCKS - 1) / (4 * GRAM_BLOCKS)) * 4;  // 400
  gram_wmma<<<GRAM_BLOCKS, 32, 0, stream>>>(s_mc, ss, rows_per_block, N);

  // scalars
  finalize_kernel<<<1, 32, 0, stream>>>(ss, scal, out + (size_t)N * NCLUST);
}